// AttendedSeqEmbedding_20023137534102
// MI455X (gfx1250) — compile-verified
//
#include <hip/hip_runtime.h>
#include <hip/hip_bf16.h>

// Problem constants (match reference)
constexpr int B = 512, L = 256, D = 256, H = 128, O = 256;
constexpr int G = 3 * H;          // 384 gate width
constexpr int BL = B * L;         // 131072 rows

typedef __attribute__((ext_vector_type(2))) float v2f;
typedef __attribute__((ext_vector_type(8))) float v8f;

// Native fp32 WMMA: D(16x16,f32) = A(16x4,f32) * B(4x16,f32) + C
__device__ __forceinline__ v8f wmma4(v2f a, v2f b, v8f c) {
    return __builtin_amdgcn_wmma_f32_16x16x4_f32(false, a, false, b,
                                                 (short)0, c, false, false);
}

// Async global->LDS copy, 16B per lane (tracked by ASYNCcnt).
// Flat pointer to __shared__ truncates to the LDS byte offset (ISA 10.2).
__device__ __forceinline__ void async_g2l_b128(void* lds, const void* gp) {
    asm volatile("global_load_async_to_lds_b128 %0, %1, off"
                 :: "v"((uint32_t)(uintptr_t)lds), "v"(gp)
                 : "memory");
}
__device__ __forceinline__ void wait_async0() {
    asm volatile("s_wait_asynccnt 0" ::: "memory");
}

__device__ __forceinline__ float sigmoid_f(float x) {
    return 1.f / (1.f + __expf(-x));
}
__device__ __forceinline__ float tanh_f(float x) {
    float e2 = __expf(-2.f * x);
    return (1.f - e2) / (1.f + e2);
}

// ---------------------------------------------------------------------------
// Stage 1: gx[row, g] = x[row,:] @ Wih^T + bih, for both directions.
// One block = one 16-row M-tile; 8 waves x 3 N-tiles x 2 dirs.
// x tile staged with async global->LDS; weights stream from L2.
// ---------------------------------------------------------------------------
__global__ __launch_bounds__(256)
void gx_kernel(const float* __restrict__ x,      // [BL, D]
               const float* __restrict__ Wf,     // [G, D]
               const float* __restrict__ bf_,    // [G]
               const float* __restrict__ Wb,     // [G, D]
               const float* __restrict__ bb_,    // [G]
               float* __restrict__ gxf,          // [BL, G]
               float* __restrict__ gxb) {
    __shared__ __align__(16) float xs[16][D + 4];
    const int mt  = blockIdx.x;
    const int tid = threadIdx.x;
    const int row0 = mt * 16;

    {   // async stage x tile: 16 rows x 256 f32, 16B chunks
        const int r = tid >> 4;
        for (int c4 = (tid & 15); c4 < D / 4; c4 += 16)
            async_g2l_b128(&xs[r][c4 * 4],
                           x + (size_t)(row0 + r) * D + c4 * 4);
    }
    wait_async0();
    __syncthreads();

    const int lane = tid & 31, wave = tid >> 5;
    const int half = lane >> 4, ln = lane & 15;
    const int n0 = wave * 48;

    v8f accf[3] = {}, accb[3] = {};
    for (int k0 = 0; k0 < D; k0 += 4) {
        const int kb = k0 + 2 * half;
        const v2f a = *(const v2f*)&xs[ln][kb];
#pragma unroll
        for (int tI = 0; tI < 3; ++tI) {
            const int n = n0 + tI * 16 + ln;
            const v2f bfv = *(const v2f*)(Wf + (size_t)n * D + kb); // B[k][n]=W[n][k]
            const v2f bbv = *(const v2f*)(Wb + (size_t)n * D + kb);
            accf[tI] = wmma4(a, bfv, accf[tI]);
            accb[tI] = wmma4(a, bbv, accb[tI]);
        }
    }
#pragma unroll
    for (int tI = 0; tI < 3; ++tI) {
        const int n = n0 + tI * 16 + ln;
        const float biasf = bf_[n], biasb = bb_[n];
#pragma unroll
        for (int j = 0; j < 8; ++j) {
            const size_t m = (size_t)(row0 + j + 8 * half);
            gxf[m * G + n] = accf[tI][j] + biasf;
            gxb[m * G + n] = accb[tI][j] + biasb;
        }
    }
}

// ---------------------------------------------------------------------------
// Stage 2: masked GRU scan. Whh B-fragments held in 192 VGPRs per lane for
// the whole scan (no per-step weight reloads on the critical path).
// Per step: async-prefetch gx tile into LDS overlapped with the WMMA GEMM.
// grid = 32 batch-tiles x 2 directions; 16 batch rows per block.
// ---------------------------------------------------------------------------
__global__ __launch_bounds__(256)
void scan_kernel(const float* __restrict__ gxf, const float* __restrict__ gxb,
                 const float* __restrict__ Whhf, const float* __restrict__ Whhb,
                 const float* __restrict__ bhhf, const float* __restrict__ bhhb,
                 const int*   __restrict__ seqlens,
                 float* __restrict__ hfo, float* __restrict__ hbo) {
    __shared__ __align__(16) float h_s [16 * (H + 4)];
    __shared__ __align__(16) float gh_s[16 * (G + 4)];
    __shared__ __align__(16) float gxs [16 * (G + 4)];
    __shared__ float bhh_s[G];
    __shared__ int sl_s[16];

    const int dir = blockIdx.x & 1;
    const int bt  = blockIdx.x >> 1;
    const int tid = threadIdx.x;
    const int b0  = bt * 16;

    const float* gx   = dir ? gxb  : gxf;
    const float* Whh  = dir ? Whhb : Whhf;
    const float* bhh  = dir ? bhhb : bhhf;
    float*       hout = dir ? hbo  : hfo;

    for (int i = tid; i < G; i += 256) bhh_s[i] = bhh[i];
    if (tid < 16) sl_s[tid] = seqlens[b0 + tid];
    for (int i = tid; i < 16 * (H + 4); i += 256) h_s[i] = 0.f;

    const int lane = tid & 31, wave = tid >> 5;
    const int half = lane >> 4, ln = lane & 15;
    const int n0 = wave * 48;

    // Preload Whh B-fragments into registers: 3 tiles x 32 k-steps x v2f.
    v2f bw[3][32];
#pragma unroll
    for (int tI = 0; tI < 3; ++tI) {
        const float* wr = Whh + (size_t)(n0 + tI * 16 + ln) * H;
#pragma unroll
        for (int kk = 0; kk < 32; ++kk)
            bw[tI][kk] = *(const v2f*)(wr + kk * 4 + 2 * half);
    }
    __syncthreads();

    const int gr  = tid >> 4;        // row for gx prefetch
    const int gc0 = tid & 15;        // chunk lane for gx prefetch

    for (int s = 0; s < L; ++s) {
        const int t = dir ? (L - 1 - s) : s;

        // async-prefetch this step's gx tile (16 x 384 f32) into LDS;
        // it lands while the WMMA GEMM below executes.
        {
            const float* src = gx + ((size_t)(b0 + gr) * L + t) * G;
            for (int c4 = gc0; c4 < G / 4; c4 += 16)
                async_g2l_b128(&gxs[gr * (G + 4) + c4 * 4], src + c4 * 4);
        }

        // gh tile [16, 384] = h[16,128] @ Whh^T; B from registers, A from LDS
        v8f acc[3] = {};
#pragma unroll
        for (int kk = 0; kk < 32; ++kk) {
            const int kb = kk * 4 + 2 * half;
            const v2f a = *(const v2f*)&h_s[ln * (H + 4) + kb];
            acc[0] = wmma4(a, bw[0][kk], acc[0]);
            acc[1] = wmma4(a, bw[1][kk], acc[1]);
            acc[2] = wmma4(a, bw[2][kk], acc[2]);
        }
#pragma unroll
        for (int tI = 0; tI < 3; ++tI) {
            const int n = n0 + tI * 16 + ln;
            const float bb = bhh_s[n];
#pragma unroll
            for (int j = 0; j < 8; ++j)
                gh_s[(j + 8 * half) * (G + 4) + n] = acc[tI][j] + bb;
        }
        wait_async0();
        __syncthreads();

        // elementwise GRU gates, float4-vectorized: 16 rows x 32 chunks
        for (int ch = tid; ch < 16 * (H / 4); ch += 256) {
            const int r  = ch >> 5;
            const int j4 = (ch & 31) * 4;
            const size_t grow = (size_t)(b0 + r) * L + t;
            const float4 xr = *(const float4*)&gxs[r * (G + 4) + j4];
            const float4 xz = *(const float4*)&gxs[r * (G + 4) + H + j4];
            const float4 xn = *(const float4*)&gxs[r * (G + 4) + 2 * H + j4];
            const float4 hr = *(const float4*)&gh_s[r * (G + 4) + j4];
            const float4 hz = *(const float4*)&gh_s[r * (G + 4) + H + j4];
            const float4 hn = *(const float4*)&gh_s[r * (G + 4) + 2 * H + j4];
            const float4 hp = *(const float4*)&h_s[r * (H + 4) + j4];
            const bool valid = t < sl_s[r];
            float4 hv;
#define GRU1(c)                                                            \
            {                                                              \
                const float rg   = sigmoid_f(xr.c + hr.c);                 \
                const float zg   = sigmoid_f(xz.c + hz.c);                 \
                const float cand = tanh_f(xn.c + rg * hn.c);               \
                const float hnew = (1.f - zg) * cand + zg * hp.c;          \
                hv.c = valid ? hnew : hp.c;                                \
            }
            GRU1(x) GRU1(y) GRU1(z) GRU1(w)
#undef GRU1
            *(float4*)&h_s[r * (H + 4) + j4] = hv;
            *(float4*)&hout[grow * H + j4]   = hv;
        }
        __syncthreads();
    }
}

// ---------------------------------------------------------------------------
// Stage 3: scores[row] = sum_o tanh(rnn[row,:] @ Wm[o,:] + bm[o]) * ctx[o]
// One block per 16-row M-tile; 8 waves x 2 N-tiles over O=256, K=2H=256.
// rnn tile staged with async global->LDS.
// ---------------------------------------------------------------------------
__global__ __launch_bounds__(256)
void score_kernel(const float* __restrict__ hf, const float* __restrict__ hb,
                  const float* __restrict__ Wm,   // [O, 2H]
                  const float* __restrict__ bm,   // [O]
                  const float* __restrict__ ctx,  // [O]
                  float* __restrict__ scores) {   // [BL]
    __shared__ __align__(16) float rn[16][2 * H + 4];
    __shared__ float part[16][O + 2];
    const int mt  = blockIdx.x;
    const int tid = threadIdx.x;
    const int row0 = mt * 16;

    {   // async stage rnn tile: row = [hf row | hb row]
        const int r = tid >> 4;
        for (int c4 = (tid & 15); c4 < (2 * H) / 4; c4 += 16) {
            const float* src = (c4 < H / 4)
                ? (hf + (size_t)(row0 + r) * H + c4 * 4)
                : (hb + (size_t)(row0 + r) * H + (c4 - H / 4) * 4);
            async_g2l_b128(&rn[r][c4 * 4], src);
        }
    }
    wait_async0();
    __syncthreads();

    const int lane = tid & 31, wave = tid >> 5;
    const int half = lane >> 4, ln = lane & 15;
    const int n0 = wave * 32;

    v8f acc[2] = {};
    for (int k0 = 0; k0 < 2 * H; k0 += 4) {
        const int kb = k0 + 2 * half;
        const v2f a = *(const v2f*)&rn[ln][kb];
#pragma unroll
        for (int tI = 0; tI < 2; ++tI) {
            const int n = n0 + tI * 16 + ln;
            const v2f bv = *(const v2f*)(Wm + (size_t)n * (2 * H) + kb);
            acc[tI] = wmma4(a, bv, acc[tI]);
        }
    }
#pragma unroll
    for (int tI = 0; tI < 2; ++tI) {
        const int n = n0 + tI * 16 + ln;
        const float bias = bm[n], cx = ctx[n];
#pragma unroll
        for (int j = 0; j < 8; ++j)
            part[j + 8 * half][n] = tanh_f(acc[tI][j] + bias) * cx;
    }
    __syncthreads();

    if (tid < 16) {
        float sum = 0.f;
        for (int n = 0; n < O; ++n) sum += part[tid][n];
        scores[row0 + tid] = sum;
    }
}

// ---------------------------------------------------------------------------
// Stage 4: per-batch masked softmax over L, then attention-pooled embedding.
// One block per batch row; 256 threads = L = 2H.
// ---------------------------------------------------------------------------
__global__ __launch_bounds__(256)
void pool_kernel(const float* __restrict__ hf, const float* __restrict__ hb,
                 const float* __restrict__ scores, const int* __restrict__ seqlens,
                 float* __restrict__ out) {       // [B, 2H]
    __shared__ float w_s[L];
    __shared__ float red[256];
    const int b   = blockIdx.x;
    const int tid = threadIdx.x;
    const int sl  = seqlens[b];

    const float sc = scores[(size_t)b * L + tid];
    const bool valid = tid < sl;

    red[tid] = valid ? sc : -3.4e38f;
    __syncthreads();
    for (int off = 128; off > 0; off >>= 1) {
        if (tid < off) red[tid] = fmaxf(red[tid], red[tid + off]);
        __syncthreads();
    }
    const float mx = red[0];
    __syncthreads();
    const float ex = valid ? __expf(sc - mx) : 0.f;
    red[tid] = ex;
    __syncthreads();
    for (int off = 128; off > 0; off >>= 1) {
        if (tid < off) red[tid] += red[tid + off];
        __syncthreads();
    }
    w_s[tid] = ex / red[0];
    __syncthreads();

    float accv = 0.f;
    if (tid < H) {
        const float* p = hf + (size_t)b * L * H + tid;
        for (int l = 0; l < L; ++l) accv += w_s[l] * p[(size_t)l * H];
    } else {
        const float* p = hb + (size_t)b * L * H + (tid - H);
        for (int l = 0; l < L; ++l) accv += w_s[l] * p[(size_t)l * H];
    }
    out[(size_t)b * (2 * H) + tid] = accv;
}

// ---------------------------------------------------------------------------
extern "C" void kernel_launch(void* const* d_in, const int* in_sizes, int n_in,
                              void* d_out, int out_size, void* d_ws, size_t ws_size,
                              hipStream_t stream) {
    const float* x     = (const float*)d_in[0];
    const int*   sl    = (const int*)  d_in[1];
    const float* Wih_f = (const float*)d_in[2];
    const float* Whh_f = (const float*)d_in[3];
    const float* bih_f = (const float*)d_in[4];
    const float* bhh_f = (const float*)d_in[5];
    const float* Wih_b = (const float*)d_in[6];
    const float* Whh_b = (const float*)d_in[7];
    const float* bih_b = (const float*)d_in[8];
    const float* bhh_b = (const float*)d_in[9];
    const float* Wm    = (const float*)d_in[10];
    const float* bm    = (const float*)d_in[11];
    const float* ctx   = (const float*)d_in[12];
    float* out = (float*)d_out;

    float* ws = (float*)d_ws;
    float* gxf = ws;    ws += (size_t)BL * G;
    float* gxb = ws;    ws += (size_t)BL * G;
    float* hf  = ws;    ws += (size_t)BL * H;
    float* hb  = ws;    ws += (size_t)BL * H;
    float* scores = ws;

    // 1) time-parallel input projection, both directions fused (8192 blocks)
    gx_kernel<<<BL / 16, 256, 0, stream>>>(x, Wih_f, bih_f, Wih_b, bih_b, gxf, gxb);

    // 2) recurrent scan, fwd+bwd concurrent (64 blocks), Whh in VGPRs
    scan_kernel<<<(B / 16) * 2, 256, 0, stream>>>(
        gxf, gxb, Whh_f, Whh_b, bhh_f, bhh_b, sl, hf, hb);

    // 3) attention MLP scores (8192 blocks)
    score_kernel<<<BL / 16, 256, 0, stream>>>(hf, hb, Wm, bm, ctx, scores);

    // 4) masked softmax + pooling (512 blocks)
    pool_kernel<<<B, 256, 0, stream>>>(hf, hb, scores, sl, out);
}